// MultiHeadAttention_33500744909308
// MI455X (gfx1250) — compile-verified
//
#include <hip/hip_runtime.h>
#include <math.h>

// MI455X (gfx1250) fused MHA, fp32 WMMA path, register-blocked.
// d_out = [attn_output (4*2048*512 f32) | attn_w (32*2048*2048 f32)]
// d_ws  = Qh | Kh | Vh | Om  (4 x 16 MB = 64 MB)

#define Bsz 4
#define Lq  2048
#define Ssz 2048
#define Dm  512
#define Hh  8
#define HD  64
#define NH  (Bsz * Hh)   // 32 heads total

typedef float v2f __attribute__((ext_vector_type(2)));
typedef float v8f __attribute__((ext_vector_type(8)));

// D = A(16x4) * B(4x16) + C(16x16), fp32, wave32.
// Per-lane fragments (ISA 7.12.2):
//   A: lane&15 = M row, regs {k0,k0+1} with k0 = 2*(lane>>4)
//   B: lane&15 = N col, regs {k0,k0+1} with k0 = 2*(lane>>4)
//   C/D: reg r holds M = r + 8*(lane>>4), N = lane&15
__device__ __forceinline__ v8f wmma_f32_16x16x4(v2f a, v2f b, v8f c) {
  return __builtin_amdgcn_wmma_f32_16x16x4_f32(
      /*neg_a=*/false, a, /*neg_b=*/false, b,
      /*c_mod=*/(short)0, c, /*reuse_a=*/false, /*reuse_b=*/false);
}

// Y = X @ W^T.  X: (M x 512) row-major, W: (512 x 512) row-major (out,in).
// Register-blocked: one wave computes a 32x64 output patch (2x4 WMMA tiles),
// so per K-step we issue 6 fragment loads for 8 WMMAs.
// mode 0: scatter Y into head-split layout (B*H, T, 64)  [Q/K/V projections]
// mode 1: plain row-major (M x 512)                      [final O projection]
#define BM 2
#define BN 4
__global__ void __launch_bounds__(32)
gemm_xwt_kernel(const float* __restrict__ X, const float* __restrict__ W,
                float* __restrict__ Y, int mode) {
  const int lane  = threadIdx.x;
  const int row16 = lane & 15;
  const int half  = lane >> 4;
  const int m0 = blockIdx.x * (16 * BM);
  const int n0 = blockIdx.y * (16 * BN);

  const float* xr[BM];
  const float* wr[BN];
#pragma unroll
  for (int i = 0; i < BM; ++i) xr[i] = X + (size_t)(m0 + 16 * i + row16) * Dm;
#pragma unroll
  for (int j = 0; j < BN; ++j) wr[j] = W + (size_t)(n0 + 16 * j + row16) * Dm;

  v8f acc[BM][BN];
#pragma unroll
  for (int i = 0; i < BM; ++i)
#pragma unroll
    for (int j = 0; j < BN; ++j) acc[i][j] = (v8f){};

#pragma unroll 2
  for (int k0 = 0; k0 < Dm; k0 += 4) {
    v2f a[BM], b[BN];
#pragma unroll
    for (int i = 0; i < BM; ++i) a[i] = *(const v2f*)(xr[i] + k0 + 2 * half);
#pragma unroll
    for (int j = 0; j < BN; ++j) b[j] = *(const v2f*)(wr[j] + k0 + 2 * half);
#pragma unroll
    for (int i = 0; i < BM; ++i)
#pragma unroll
      for (int j = 0; j < BN; ++j)
        acc[i][j] = wmma_f32_16x16x4(a[i], b[j], acc[i][j]);
  }

#pragma unroll
  for (int i = 0; i < BM; ++i) {
#pragma unroll
    for (int j = 0; j < BN; ++j) {
#pragma unroll
      for (int r = 0; r < 8; ++r) {
        const int mr = m0 + 16 * i + r + 8 * half;
        const int nc = n0 + 16 * j + row16;
        const float v = acc[i][j][r];
        if (mode == 0) {
          const int b_ = mr / Lq, t = mr % Lq;
          const int h = nc / HD, hd = nc % HD;
          Y[(((size_t)(b_ * Hh + h) * Lq) + t) * HD + hd] = v;
        } else {
          Y[(size_t)mr * Dm + nc] = v;
        }
      }
    }
  }
}

// One workgroup per (head, 32 query rows): scores (WMMA, K=64) -> LDS,
// softmax in LDS (attn_w written once), then O = P @ V (WMMA, split-K).
#define ROWS    32
#define WAVES   16
#define THREADS (WAVES * 32)
#define PART_F  (WAVES * 512)          // split-K partials: 16 waves x 2 tiles x 256

__global__ void __launch_bounds__(THREADS)
attn_kernel(const float* __restrict__ Qh, const float* __restrict__ Kh,
            const float* __restrict__ Vh, const unsigned char* __restrict__ mask,
            float* __restrict__ attn_w, float* __restrict__ Om) {
  extern __shared__ float smem[];
  float* sc   = smem;                       // ROWS x Ssz scores (256 KB)
  float* part = smem + (size_t)ROWS * Ssz;  // split-K partials (32 KB)

  const int nh   = blockIdx.y;       // head 0..31
  const int lblk = blockIdx.x;       // row block 0..63
  const int tid  = threadIdx.x;
  const int w    = tid >> 5;
  const int lane = tid & 31;
  const int row16 = lane & 15;
  const int half  = lane >> 4;
  const int l0 = lblk * ROWS;

  const float* Qb = Qh + (size_t)nh * Lq * HD;
  const float* Kb = Kh + (size_t)nh * Ssz * HD;
  const float* Vb = Vh + (size_t)nh * Ssz * HD;

  // ---- Phase 1: scores = (Q K^T) / 8, masked, into LDS --------------------
  // 64 units of (1 row-tile x 4 col-tiles); Q fragment reused across 4 WMMAs.
  for (int u = w; u < 64; u += WAVES) {
    const int tr = u >> 5;           // 0..1
    const int cb = u & 31;           // col block of 4 tiles
    const int t0 = l0 + tr * 16;
    const int s0 = cb * 64;

    const float* qrow = Qb + (size_t)(t0 + row16) * HD;
    const float* kr[4];
#pragma unroll
    for (int j = 0; j < 4; ++j)
      kr[j] = Kb + (size_t)(s0 + 16 * j + row16) * HD;  // B[k,s] = K[s,k]

    v8f acc[4];
#pragma unroll
    for (int j = 0; j < 4; ++j) acc[j] = (v8f){};

#pragma unroll
    for (int k0 = 0; k0 < HD; k0 += 4) {
      v2f a = *(const v2f*)(qrow + k0 + 2 * half);
#pragma unroll
      for (int j = 0; j < 4; ++j) {
        v2f b = *(const v2f*)(kr[j] + k0 + 2 * half);
        acc[j] = wmma_f32_16x16x4(a, b, acc[j]);
      }
    }
#pragma unroll
    for (int j = 0; j < 4; ++j) {
#pragma unroll
      for (int r = 0; r < 8; ++r) {
        const int rl = tr * 16 + r + 8 * half;        // local row 0..31
        const int s  = s0 + 16 * j + row16;
        float v = acc[j][r] * 0.125f;                 // 1/sqrt(64)
        if (mask[(size_t)(l0 + rl) * Ssz + s]) v = -1e30f;
        sc[(size_t)rl * Ssz + s] = v;
      }
    }
  }
  __syncthreads();

  // ---- Phase 2: softmax rows; wave w: rows 2w (lanes 0-15), 2w+1 (16-31) --
  {
    const int rl = 2 * w + half;
    float* srow = sc + (size_t)rl * Ssz;
    float mx = -INFINITY;
    for (int i = 0; i < Ssz / 16; ++i)
      mx = fmaxf(mx, srow[row16 + 16 * i]);
    for (int m = 8; m >= 1; m >>= 1) mx = fmaxf(mx, __shfl_xor(mx, m, 16));

    float sum = 0.f;
    for (int i = 0; i < Ssz / 16; ++i) {
      const float p = __expf(srow[row16 + 16 * i] - mx);
      srow[row16 + 16 * i] = p;
      sum += p;
    }
    for (int m = 8; m >= 1; m >>= 1) sum += __shfl_xor(sum, m, 16);
    const float inv = 1.0f / sum;

    float* arow = attn_w + ((size_t)nh * Lq + (l0 + rl)) * Ssz;
    for (int i = 0; i < Ssz / 16; ++i) {
      const float p = srow[row16 + 16 * i] * inv;
      srow[row16 + 16 * i] = p;          // keep normalized P for phase 3
      arow[row16 + 16 * i] = p;          // single coalesced attn_w write
    }
  }
  __syncthreads();

  // ---- Phase 3: O(32x64) = P(32x2048) @ V(2048x64) ------------------------
  // Wave (kq, c): K-quarter kq in 0..3, col-tile c in 0..3.
  // V fragment shared across both row tiles: 2 WMMAs per B load.
  {
    const int c  = w & 3;
    const int kq = w >> 2;
    const int n0 = c * 16;

    const float* prow0 = sc + (size_t)(0 + row16) * Ssz;
    const float* prow1 = sc + (size_t)(16 + row16) * Ssz;
    const int kbeg = kq * (Ssz / 4);

    v8f acc0 = {}, acc1 = {};
    for (int k0 = kbeg; k0 < kbeg + Ssz / 4; k0 += 4) {
      v2f a0 = *(const v2f*)(prow0 + k0 + 2 * half);
      v2f a1 = *(const v2f*)(prow1 + k0 + 2 * half);
      v2f b;
      b.x = Vb[(size_t)(k0 + 2 * half + 0) * HD + n0 + row16];
      b.y = Vb[(size_t)(k0 + 2 * half + 1) * HD + n0 + row16];
      acc0 = wmma_f32_16x16x4(a0, b, acc0);
      acc1 = wmma_f32_16x16x4(a1, b, acc1);
    }
#pragma unroll
    for (int r = 0; r < 8; ++r) {
      part[w * 512 + 0 * 256 + r * 32 + lane] = acc0[r];
      part[w * 512 + 1 * 256 + r * 32 + lane] = acc1[r];
    }
  }
  __syncthreads();

  if (w < 4) {                      // wave w reduces col-tile c = w
    const int c  = w;
    const int n0 = c * 16;
    const int bidx = nh / Hh, hidx = nh % Hh;
#pragma unroll
    for (int i = 0; i < 2; ++i) {   // row tiles
#pragma unroll
      for (int r = 0; r < 8; ++r) {
        float v = 0.f;
#pragma unroll
        for (int q = 0; q < 4; ++q)
          v += part[(size_t)(q * 4 + c) * 512 + i * 256 + r * 32 + lane];
        const int rl = i * 16 + r + 8 * half;
        const int lg = l0 + rl;
        // merged (B, L, D) layout for the final projection
        Om[((size_t)(bidx * Lq + lg)) * Dm + hidx * HD + n0 + row16] = v;
      }
    }
  }
}

extern "C" void kernel_launch(void* const* d_in, const int* in_sizes, int n_in,
                              void* d_out, int out_size, void* d_ws, size_t ws_size,
                              hipStream_t stream) {
  const float* queries = (const float*)d_in[0];
  const float* keys    = (const float*)d_in[1];
  const float* values  = (const float*)d_in[2];
  const unsigned char* mask = (const unsigned char*)d_in[3];  // bool (L,S)
  const float* W_q = (const float*)d_in[4];
  const float* W_k = (const float*)d_in[5];
  const float* W_v = (const float*)d_in[6];
  const float* W_o = (const float*)d_in[7];

  float* attn_out = (float*)d_out;                               // (B,L,D)
  float* attn_w   = attn_out + (size_t)Bsz * Lq * Dm;            // (B*H,L,S)

  const size_t per = (size_t)NH * Lq * HD;  // 4,194,304 floats = 16 MB
  float* Qh = (float*)d_ws;
  float* Kh = Qh + per;
  float* Vh = Kh + per;
  float* Om = Vh + per;                     // total 64 MB of d_ws

  const dim3 gproj((Bsz * Lq) / (16 * BM), Dm / (16 * BN));  // (256, 8)
  gemm_xwt_kernel<<<gproj, 32, 0, stream>>>(queries, W_q, Qh, 0);
  gemm_xwt_kernel<<<gproj, 32, 0, stream>>>(keys,    W_k, Kh, 0);
  gemm_xwt_kernel<<<gproj, 32, 0, stream>>>(values,  W_v, Vh, 0);

  const size_t smem_bytes =
      ((size_t)ROWS * Ssz + (size_t)PART_F) * sizeof(float);  // 288 KB
  const dim3 gattn(Lq / ROWS, NH);  // (64, 32)
  attn_kernel<<<gattn, THREADS, smem_bytes, stream>>>(Qh, Kh, Vh, mask,
                                                      attn_w, Om);

  gemm_xwt_kernel<<<gproj, 32, 0, stream>>>(Om, W_o, attn_out, 1);
}